// SelfCorrelationPercPooling_19215683682884
// MI455X (gfx1250) — compile-verified
//
#include <hip/hip_runtime.h>
#include <hip/hip_bf16.h>

typedef __attribute__((ext_vector_type(2))) float v2f;
typedef __attribute__((ext_vector_type(8))) float v8f;
typedef __attribute__((ext_vector_type(4))) unsigned int v4u;
typedef __attribute__((ext_vector_type(8))) int v8i;
typedef __attribute__((ext_vector_type(4))) int v4i;

#define NMAPS 1024
#define FDIM  512
#define TILE  128      // macro tile (M and N)
#define KTILE 32
#define LDSP  36       // padded row pitch in floats (144B: 16B aligned, conflict-free)
#define NCHUNK (FDIM / KTILE)   // 16

// ---------------------------------------------------------------------------
// TDM: DMA one 128-row x 32-float panel (row-major, row stride 512 floats)
// from global into LDS with a 4-DWORD pad after every 32 DWORDs -> 36f pitch.
// D# per CDNA5 ISA ch.8: group0 {count,lds_addr,global_addr,type=2},
// group1 {data_size=4B, pad 32dw+4dw, dims/strides}, groups 2/3 = 0 (2D).
// Toolchain here is the 6-arg clang-23 builtin form.
// ---------------------------------------------------------------------------
__device__ __forceinline__ void tdm_load_panel(const float* gsrc, float* ldsdst) {
  const unsigned long long ga = (unsigned long long)(uintptr_t)gsrc;
  const unsigned int la = (unsigned int)(uintptr_t)ldsdst;  // LDS byte offset

  v4u g0;
  g0.x = 1u;                                       // count=1 (valid user D#)
  g0.y = la;                                       // lds_addr
  g0.z = (unsigned int)(ga & 0xffffffffull);       // global_addr[31:0]
  g0.w = (unsigned int)(ga >> 32) | (2u << 30);    // global_addr[56:32] | type=2

  v8i g1;
  g1[0] = (int)((2u << 16)        // data_size = 4 bytes
              | (1u << 20)        // pad_enable
              | (4u << 22)        // pad_interval: 32 DWORDs
              | (3u << 25));      // pad_amount:   4 DWORDs
  g1[1] = (int)((FDIM & 0xffffu) << 16);            // tensor_dim0=512 (lo16)
  g1[2] = (int)((FDIM >> 16) | ((NMAPS & 0xffffu) << 16)); // dim0 hi | dim1 lo
  g1[3] = (int)((NMAPS >> 16) | (KTILE << 16));     // dim1 hi | tile_dim0=32
  g1[4] = (int)(TILE);                              // tile_dim1=128, tile_dim2=0
  g1[5] = (int)(FDIM);                              // tensor_dim0_stride=512 (lo32)
  g1[6] = 0;                                        // stride hi | dim1_stride lo
  g1[7] = 0;

  const v4i gz4 = {0, 0, 0, 0};
  const v8i gz8 = {0, 0, 0, 0, 0, 0, 0, 0};
  __builtin_amdgcn_tensor_load_to_lds(g0, g1, gz4, gz4, gz8, 0);
}

// ---------------------------------------------------------------------------
// Kernel 1: corr[b] = X[b] * X[b]^T / 512   (X[b] is 1024 x 512, row-major)
// grid: (64 tiles, nBatchInChunk), block: 256 threads (8 wave32)
// Double-buffered TDM staging overlapped with f32 WMMA compute.
// ---------------------------------------------------------------------------
__global__ __launch_bounds__(256)
void corr_gemm_kernel(const float* __restrict__ x, float* __restrict__ corr,
                      int batch0) {
  const int b  = blockIdx.y;            // batch within chunk
  const int gb = batch0 + b;            // global batch
  const int tm = (blockIdx.x & 7) * TILE;
  const int tn = (blockIdx.x >> 3) * TILE;

  const float* __restrict__ X = x + (size_t)gb * NMAPS * FDIM;
  float* __restrict__ C = corr + (size_t)b * NMAPS * NMAPS;

  __shared__ float As[2][TILE][LDSP];
  __shared__ float Bs[2][TILE][LDSP];

  const int tid  = threadIdx.x;
  const int lane = tid & 31;
  const int wave = tid >> 5;
  const int lr   = lane & 15;          // fragment row within 16-tile
  const int lk2  = (lane >> 4) << 1;   // K sub-offset: 0 or 2
  const int rowA = (wave & 3) * 32;    // wave's row offset within macro tile
  const int colB = (wave >> 2) * 64;   // wave's col offset within macro tile

  v8f acc[2][4];
#pragma unroll
  for (int i = 0; i < 2; ++i)
#pragma unroll
    for (int j = 0; j < 4; ++j)
      acc[i][j] = (v8f){0.f, 0.f, 0.f, 0.f, 0.f, 0.f, 0.f, 0.f};

  // prologue: wave 0 issues the chunk-0 DMAs
  if (wave == 0) {
    tdm_load_panel(X + (size_t)tm * FDIM, &As[0][0][0]);
    tdm_load_panel(X + (size_t)tn * FDIM, &Bs[0][0][0]);
  }

  for (int c = 0; c < NCHUNK; ++c) {
    const int cur = c & 1;
    if (wave == 0) {
      if (c + 1 < NCHUNK) {
        const int k1 = (c + 1) * KTILE;
        tdm_load_panel(X + (size_t)tm * FDIM + k1, &As[cur ^ 1][0][0]);
        tdm_load_panel(X + (size_t)tn * FDIM + k1, &Bs[cur ^ 1][0][0]);
        __builtin_amdgcn_s_wait_tensorcnt(2);  // chunk c done; c+1 in flight
      } else {
        __builtin_amdgcn_s_wait_tensorcnt(0);  // drain last chunk
      }
    }
    __syncthreads();   // buffer `cur` ready for all waves

#pragma unroll
    for (int kk = 0; kk < KTILE; kk += 4) {
      v2f af[2], bf[4];
#pragma unroll
      for (int i = 0; i < 2; ++i)
        af[i] = *(const v2f*)&As[cur][rowA + i * 16 + lr][kk + lk2];
#pragma unroll
      for (int j = 0; j < 4; ++j)
        bf[j] = *(const v2f*)&Bs[cur][colB + j * 16 + lr][kk + lk2];
#pragma unroll
      for (int i = 0; i < 2; ++i)
#pragma unroll
        for (int j = 0; j < 4; ++j)
          acc[i][j] = __builtin_amdgcn_wmma_f32_16x16x4_f32(
              false, af[i], false, bf[j], (short)0, acc[i][j], false, false);
    }
    __syncthreads();   // all waves done reading `cur` before TDM refills it
  }

  // store: C/D layout -> VGPR v: lanes0-15 M=v, lanes16-31 M=v+8; N = lane&15
  const float invF = 1.0f / (float)FDIM;
  const int mhi = (lane >> 4) << 3;
#pragma unroll
  for (int i = 0; i < 2; ++i)
#pragma unroll
    for (int j = 0; j < 4; ++j)
#pragma unroll
      for (int v = 0; v < 8; ++v) {
        const int row = tm + rowA + i * 16 + v + mhi;
        const int col = tn + colB + j * 16 + (lane & 15);
        C[(size_t)row * NMAPS + col] = acc[i][j][v] * invF;
      }
}

// ---------------------------------------------------------------------------
// Kernel 2: per row of corr: descending bitonic sort of 1024 f32 in LDS,
// then gather 256 percentile ranks.  grid: (1024 rows, nBatchInChunk)
// ---------------------------------------------------------------------------
__global__ __launch_bounds__(256)
void sort_rank_kernel(const float* __restrict__ corr, float* __restrict__ out,
                      int batch0) {
  __shared__ float s[NMAPS];
  const int b   = blockIdx.y;
  const int row = blockIdx.x;
  const int tid = threadIdx.x;

  const float* __restrict__ src = corr + ((size_t)b * NMAPS + row) * NMAPS;
  ((float4*)s)[tid] = ((const float4*)src)[tid];   // 1024 floats / 256 thr
  __syncthreads();

  // bitonic sort, descending
  for (int k = 2; k <= NMAPS; k <<= 1) {
    for (int j = k >> 1; j > 0; j >>= 1) {
#pragma unroll
      for (int t = tid; t < NMAPS; t += 256) {
        const int ixj = t ^ j;
        if (ixj > t) {
          const float a = s[t];
          const float c = s[ixj];
          const bool desc = ((t & k) == 0);
          if (desc ? (a < c) : (a > c)) { s[t] = c; s[ixj] = a; }
        }
      }
      __syncthreads();
    }
  }

  // ranks = round(linspace(1.0, 1023.0, 256)), round-half-even like jnp.round
  const int r = (int)rint(1.0 + (double)tid * (1022.0 / 255.0));
  out[(((size_t)(batch0 + b) * NMAPS + row) << 8) + tid] = s[r];
}

// ---------------------------------------------------------------------------
extern "C" void kernel_launch(void* const* d_in, const int* in_sizes, int n_in,
                              void* d_out, int out_size, void* d_ws,
                              size_t ws_size, hipStream_t stream) {
  const float* x = (const float*)d_in[0];
  float* out  = (float*)d_out;
  float* corr = (float*)d_ws;

  const size_t perBatch = (size_t)NMAPS * NMAPS * sizeof(float);  // 4 MB
  int chunk = (int)(ws_size / perBatch);
  if (chunk < 1)  chunk = 1;   // assume ws holds at least one batch
  if (chunk > 32) chunk = 32;

  for (int b0 = 0; b0 < 32; b0 += chunk) {
    const int nb = (32 - b0 < chunk) ? (32 - b0) : chunk;
    dim3 g1(64, nb);
    corr_gemm_kernel<<<g1, 256, 0, stream>>>(x, corr, b0);
    dim3 g2(NMAPS, nb);
    sort_rank_kernel<<<g2, 256, 0, stream>>>(corr, out, b0);
  }
}